// Point2PointLoss_27642409517714
// MI455X (gfx1250) — compile-verified
//
#include <hip/hip_runtime.h>
#include <hip/hip_bf16.h>

// B=4, M=1024, N=32768.
//   min_n ||k-p||^2 = ||k||^2 + min_n( ||p||^2 - 2 k.p )
// Inner term via V_WMMA_F32_16X16X4_F32:
//   A[m] = [-2kx, -2ky, -2kz, 1]  (16x4 f32; lanes 0-15 K0/K1, lanes 16-31 K2/K3)
//   B[n] = [ px,   py,   pz, p2]  (4x16 f32; same half split)
// B operand fully pre-packed in WMMA lane order in LDS -> inner loop is
// ds_load_2addr + wmma + 4x v_min3 per tile. sqrt/max/+k2 hoisted out.

typedef float v2f __attribute__((ext_vector_type(2)));
typedef float v8f __attribute__((ext_vector_type(8)));

#define PM   1024
#define PN   32768
#define NSL  4096            // N-slice per block
#define NT   (NSL / 16)      // 256 tiles per slice
#define NWAVE 8              // waves per block; each wave owns one 16-row M-tile

__global__ __launch_bounds__(256) void
p2p_min_kernel(const float* __restrict__ kp,   // [B,3,M]
               const float* __restrict__ pc,   // [B,3,N]
               float* __restrict__ rowmin)     // [256 blocks][8 waves][16 rows]
{
    // Packed B operands: per tile t, 64 floats:
    //  [ 0..15] px   (b0, lanes 0-15)    [16..31] pz  (b0, lanes 16-31)
    //  [32..47] py   (b1, lanes 0-15)    [48..63] p2  (b1, lanes 16-31)
    __shared__ float pak[NT * 64];       // 64 KB

    const int bi = blockIdx.x;           // bi = ((b*8 + mg)*8 + ns)
    const int ns = bi & 7;
    const int mg = (bi >> 3) & 7;
    const int b  = bi >> 6;
    const int n0 = ns * NSL;

    const float* pcb = pc + (size_t)b * 3 * PN;
    const float* kpb = kp + (size_t)b * 3 * PM;

    // ---- Stage & pack this N-slice into LDS (one tile per thread) ----
    {
        const int t = threadIdx.x;       // 256 threads == 256 tiles
        const float4* px4 = (const float4*)(pcb + n0 + t * 16);
        const float4* py4 = (const float4*)(pcb + PN + n0 + t * 16);
        const float4* pz4 = (const float4*)(pcb + 2 * PN + n0 + t * 16);
        float4* dst = (float4*)&pak[t * 64];
#pragma unroll
        for (int q = 0; q < 4; ++q) {
            const float4 x = px4[q];
            const float4 y = py4[q];
            const float4 z = pz4[q];
            float4 p2;
            p2.x = __builtin_fmaf(x.x, x.x, __builtin_fmaf(y.x, y.x, z.x * z.x));
            p2.y = __builtin_fmaf(x.y, x.y, __builtin_fmaf(y.y, y.y, z.y * z.y));
            p2.z = __builtin_fmaf(x.z, x.z, __builtin_fmaf(y.z, y.z, z.z * z.z));
            p2.w = __builtin_fmaf(x.w, x.w, __builtin_fmaf(y.w, y.w, z.w * z.w));
            dst[q]      = x;   // px  -> b0 lanes 0-15
            dst[4 + q]  = z;   // pz  -> b0 lanes 16-31
            dst[8 + q]  = y;   // py  -> b1 lanes 0-15
            dst[12 + q] = p2;  // p2  -> b1 lanes 16-31
        }
    }
    __syncthreads();

    const int lane = threadIdx.x & 31;
    const int w    = threadIdx.x >> 5;   // wave id -> M-tile
    const int half = lane >> 4;
    const int lm   = lane & 15;
    const int mbase = (mg * NWAVE + w) * 16;

    // ---- A operand (constant across the whole N scan) ----
    v2f A;
    {
        const int m = mbase + lm;
        if (half == 0) {
            A.x = -2.0f * kpb[m];
            A.y = -2.0f * kpb[PM + m];
        } else {
            A.x = -2.0f * kpb[2 * PM + m];
            A.y = 1.0f;                  // K=3 slot multiplies ||p||^2
        }
    }

    v8f rmin;
#pragma unroll
    for (int v = 0; v < 8; ++v) rmin[v] = 3.0e38f;

    // ---- Hot loop: all LDS-fed, 1 WMMA + 4 min3 per 256 pairs ----
#pragma unroll 4
    for (int t = 0; t < NT; ++t) {
        const int base = t * 64 + lane;
        const float b0 = pak[base];          // px | pz
        const float b1 = pak[base + 32];     // py | p2
        v2f Bv; Bv.x = b0; Bv.y = b1;
        v8f c = {};
        v8f d = __builtin_amdgcn_wmma_f32_16x16x4_f32(
            false, A, false, Bv, (short)0, c, false, false);
#pragma unroll
        for (int v = 0; v < 8; ++v) rmin[v] = fminf(rmin[v], d[v]);
    }

    // ---- Min across the 16 columns held inside each half-wave ----
#pragma unroll
    for (int mask = 8; mask >= 1; mask >>= 1) {
#pragma unroll
        for (int v = 0; v < 8; ++v) {
            const float o = __shfl_xor(rmin[v], mask, 32);
            rmin[v] = fminf(rmin[v], o);
        }
    }

    // Lane 0 holds rows 0-7, lane 16 holds rows 8-15 of this wave's M-tile.
    if (lm == 0) {
        float* dst = rowmin + ((size_t)bi * NWAVE + w) * 16 + half * 8;
#pragma unroll
        for (int v = 0; v < 8; ++v) dst[v] = rmin[v];
    }
}

// Deterministic finish: min over the 8 N-slices, +||k||^2, sqrt, mean.
__global__ __launch_bounds__(256) void
p2p_reduce_kernel(const float* __restrict__ rowmin,
                  const float* __restrict__ kp,
                  float* __restrict__ out)
{
    __shared__ float sbuf[256];
    float s = 0.0f;
#pragma unroll
    for (int i = 0; i < 16; ++i) {
        const int g  = threadIdx.x + i * 256;   // row 0..4095
        const int b  = g >> 10;
        const int m  = g & 1023;
        const int mt = m >> 4;
        const int mg = mt >> 3;
        const int w  = mt & 7;
        const int r  = m & 15;
        // slot(ns) = (((b*8+mg)*8 + ns)*8 + w)*16 + r ; ns-stride = 128
        const float* p = rowmin + ((size_t)((b * 8 + mg) * 8) * 8 + w) * 16 + r;
        float v = p[0];
#pragma unroll
        for (int nsl = 1; nsl < 8; ++nsl) v = fminf(v, p[nsl * 128]);
        const float kx = kp[(size_t)b * 3 * PM + m];
        const float ky = kp[(size_t)b * 3 * PM + PM + m];
        const float kz = kp[(size_t)b * 3 * PM + 2 * PM + m];
        const float k2 = __builtin_fmaf(kx, kx, __builtin_fmaf(ky, ky, kz * kz));
        s += __builtin_sqrtf(fmaxf(v + k2, 0.0f));
    }
    sbuf[threadIdx.x] = s;
    __syncthreads();
    for (int off = 128; off > 0; off >>= 1) {
        if (threadIdx.x < off) sbuf[threadIdx.x] += sbuf[threadIdx.x + off];
        __syncthreads();
    }
    if (threadIdx.x == 0) out[0] = sbuf[0] * (1.0f / 4096.0f);
}

extern "C" void kernel_launch(void* const* d_in, const int* in_sizes, int n_in,
                              void* d_out, int out_size, void* d_ws, size_t ws_size,
                              hipStream_t stream)
{
    const float* kp = (const float*)d_in[0];   // keypoints [4,3,1024]
    const float* pc = (const float*)d_in[1];   // pc        [4,3,32768]
    float* rowmin = (float*)d_ws;              // 256*8*16 floats = 128 KB scratch

    p2p_min_kernel<<<dim3(256), dim3(256), 0, stream>>>(kp, pc, rowmin);
    p2p_reduce_kernel<<<dim3(1), dim3(256), 0, stream>>>(rowmin, kp, (float*)d_out);
}